// GOLCNN_29738353557601
// MI455X (gfx1250) — compile-verified
//
#include <hip/hip_runtime.h>
#include <cstdint>

// Game of Life step: y = conv3x3(x, [[2,2,2],[2,1,2],[2,2,2]], wrap) + bias
//                    out = (y>=4.5) - (y>=7.5)
// Identity used: conv == 2*boxsum3x3 - center  (exact in fp32, integer sums <= 17)

#define GOL_H 2048
#define GOL_W 2048
#define TILE_W 128
#define TILE_H 32
#define NLROWS (TILE_H + 2)   // 34 staged rows (top/bottom halo)
#define LDSP   136            // LDS row stride in floats (16B-aligns interior b128 writes)
#define HALO_L 3              // LDS column holding global col0-1

typedef float v4f __attribute__((ext_vector_type(4)));

// ---- CDNA5 async global->LDS copies (ASYNCcnt path, CDNA5 ISA) ----
__device__ __forceinline__ void async_load_b128(uint32_t lds_off, const float* gaddr) {
  asm volatile("global_load_async_to_lds_b128 %0, %1, off"
               :: "v"(lds_off), "v"(gaddr) : "memory");
}
__device__ __forceinline__ void async_load_b32(uint32_t lds_off, const float* gaddr) {
  asm volatile("global_load_async_to_lds_b32 %0, %1, off"
               :: "v"(lds_off), "v"(gaddr) : "memory");
}
__device__ __forceinline__ void wait_async0() {
  asm volatile("s_wait_asynccnt 0" ::: "memory");
}

__global__ __launch_bounds__(256) void GOLCNN_29738353557601_kernel(
    const float* __restrict__ x, const float* __restrict__ bias_p,
    float* __restrict__ out) {
  __shared__ float tile[NLROWS * LDSP];

  const int tid  = threadIdx.x;
  const int col0 = blockIdx.x * TILE_W;   // multiple of 128 -> b128-aligned global loads
  const int row0 = blockIdx.y * TILE_H;
  const size_t img = (size_t)blockIdx.z * ((size_t)GOL_H * GOL_W);

  // ---- Phase 1: stage (TILE_H+2) x (TILE_W+2) haloed tile into LDS, async ----
  // Interior: 34 rows x 32 x b128 (128 floats/row), row index wraps on the torus.
  for (int t = tid; t < NLROWS * (TILE_W / 4); t += 256) {
    const int lr = t >> 5;                    // staged row 0..33
    const int lq = t & 31;                    // which 4-float group
    const int gr = (row0 + lr - 1) & (GOL_H - 1);
    const float* gp = x + img + (size_t)gr * GOL_W + (size_t)(col0 + lq * 4);
    const uint32_t loff =
        (uint32_t)(uintptr_t)&tile[lr * LDSP + (HALO_L + 1) + lq * 4];
    async_load_b128(loff, gp);
  }
  // Halo columns (wrapped): 34 rows x {left,right}, one b32 each.
  if (tid < 2 * NLROWS) {
    const int lr   = tid >> 1;
    const int side = tid & 1;
    const int gr = (row0 + lr - 1) & (GOL_H - 1);
    const int gc = side ? ((col0 + TILE_W) & (GOL_W - 1))
                        : ((col0 - 1) & (GOL_W - 1));
    const int lc = side ? (HALO_L + 1 + TILE_W) : HALO_L;
    const float* gp = x + img + (size_t)gr * GOL_W + (size_t)gc;
    const uint32_t loff = (uint32_t)(uintptr_t)&tile[lr * LDSP + lc];
    async_load_b32(loff, gp);
  }

  wait_async0();      // this wave's async LDS writes are complete
  __syncthreads();    // all waves' staged data visible

  // ---- Phase 2: each thread computes a 4x4 output patch from LDS ----
  const float bias = bias_p[0];
  const int ox = (tid & 31) * 4;  // 32 groups * 4 cols = 128
  const int oy = (tid >> 5) * 4;  //  8 groups * 4 rows = 32
  const int base = oy * LDSP + ox + HALO_L;  // lds col of (global col0+ox-1)

  float c0[6], c1[6], c2[6];      // three staged rows, 6 columns (4 outputs + halo)
#pragma unroll
  for (int j = 0; j < 6; ++j) {
    c0[j] = tile[base + j];
    c1[j] = tile[base + LDSP + j];
    c2[j] = tile[base + 2 * LDSP + j];
  }

#pragma unroll
  for (int r = 0; r < 4; ++r) {
    float vs[6];
#pragma unroll
    for (int j = 0; j < 6; ++j) vs[j] = c0[j] + c1[j] + c2[j];  // vertical 3-sums

    v4f o;
#pragma unroll
    for (int c = 0; c < 4; ++c) {
      const float S = vs[c] + vs[c + 1] + vs[c + 2];  // 3x3 box sum
      const float y = 2.0f * S - c1[c + 1] + bias;    // conv + bias (exact)
      o[c] = ((y >= 4.5f) ? 1.0f : 0.0f) - ((y >= 7.5f) ? 1.0f : 0.0f);
    }
    float* op = out + img + (size_t)(row0 + oy + r) * GOL_W + (size_t)(col0 + ox);
    __builtin_nontemporal_store(o, (v4f*)op);  // streamed output, keep L2 for input halos

    if (r < 3) {
#pragma unroll
      for (int j = 0; j < 6; ++j) {
        c0[j] = c1[j];
        c1[j] = c2[j];
        c2[j] = tile[base + (r + 3) * LDSP + j];
      }
    }
  }
}

extern "C" void kernel_launch(void* const* d_in, const int* in_sizes, int n_in,
                              void* d_out, int out_size, void* d_ws, size_t ws_size,
                              hipStream_t stream) {
  const float* x    = (const float*)d_in[0];  // [B,1,2048,2048] fp32
  // d_in[1] is the fixed GoL stencil (structure hardcoded via 2*boxsum - center)
  const float* bias = (const float*)d_in[2];  // [1] fp32
  float* out = (float*)d_out;

  const int B = in_sizes[0] / (GOL_H * GOL_W);
  dim3 grid(GOL_W / TILE_W, GOL_H / TILE_H, B);
  GOLCNN_29738353557601_kernel<<<grid, 256, 0, stream>>>(x, bias, out);
}